// FarthestPointSampler_12584254178061
// MI455X (gfx1250) — compile-verified
//
#include <hip/hip_runtime.h>
#include <hip/hip_bf16.h>

// ---------------------------------------------------------------------------
// Problem constants (from reference)
// ---------------------------------------------------------------------------
#define B_SZ   8
#define D_FEAT 64
#define NPTS   16384
#define MPTS   2048      // N_POINTS
#define KNN    16        // K
#define FPS_T  1024      // threads per FPS block
#define FPS_R  (NPTS / FPS_T)  // 16 points per thread

typedef __attribute__((ext_vector_type(2))) float v2f;
typedef __attribute__((ext_vector_type(8))) float v8f;

#define BIGF 3.402823466e38f

// ---------------------------------------------------------------------------
// Kernel P: p2[b][n] = |xyz[b,:,n]|^2   (rank key helper; the s2 row-constant
// cannot change per-row top-k ordering, so it is dropped entirely)
// ---------------------------------------------------------------------------
__global__ __launch_bounds__(256) void p2_kernel(const float* __restrict__ xyz,
                                                 float* __restrict__ p2) {
    int i = blockIdx.x * 256 + threadIdx.x;          // over B*N
    if (i >= B_SZ * NPTS) return;
    int b = i >> 14;           // / NPTS
    int n = i & (NPTS - 1);
    const float* xb = xyz + (size_t)b * 3 * NPTS;
    float X = xb[n], Y = xb[NPTS + n], Z = xb[2 * NPTS + n];
    p2[i] = X * X + Y * Y + Z * Z;
}

// ---------------------------------------------------------------------------
// Kernel A: farthest point sampling. One block per batch; the 16 points/thread
// and running min-distances live entirely in VGPRs; LDS only for the argmax
// reduction + winner broadcast (3 barriers per iteration).
// ---------------------------------------------------------------------------
__global__ __launch_bounds__(FPS_T) void fps_kernel(const float* __restrict__ xyz,
                                                    float* __restrict__ out_ind,   // [B][M] as float
                                                    float* __restrict__ ws_sxyz) { // [B][3][M]
    const int b = blockIdx.x;
    const int tid = threadIdx.x;
    const int lane = tid & 31, wid = tid >> 5;
    const float* xb = xyz + (size_t)b * 3 * NPTS;

    float px[FPS_R], py[FPS_R], pz[FPS_R], dmin[FPS_R];
#pragma unroll
    for (int r = 0; r < FPS_R; ++r) {
        int n = r * FPS_T + tid;                       // coalesced
        px[r] = xb[n];
        py[r] = xb[NPTS + n];
        pz[r] = xb[2 * NPTS + n];
        dmin[r] = BIGF;
    }

    __shared__ float red_d[32];
    __shared__ int   red_i[32];
    __shared__ float bx, by, bz;
    __shared__ int   bwin;

    if (tid == 0) {
        bwin = 0;
        bx = xb[0]; by = xb[NPTS]; bz = xb[2 * NPTS];
        out_ind[(size_t)b * MPTS] = 0.0f;              // deterministic start idx 0
        ws_sxyz[((size_t)b * 3 + 0) * MPTS] = xb[0];
        ws_sxyz[((size_t)b * 3 + 1) * MPTS] = xb[NPTS];
        ws_sxyz[((size_t)b * 3 + 2) * MPTS] = xb[2 * NPTS];
    }
    __syncthreads();

    for (int i = 1; i < MPTS; ++i) {
        float lx = bx, ly = by, lz = bz;
        float best = -1.0f;
        int bestn = 0;
#pragma unroll
        for (int r = 0; r < FPS_R; ++r) {
            float dx = px[r] - lx, dy = py[r] - ly, dz = pz[r] - lz;
            float d = dx * dx + dy * dy + dz * dz;
            dmin[r] = fminf(dmin[r], d);
            if (dmin[r] > best) { best = dmin[r]; bestn = r * FPS_T + tid; }
        }
        // intra-wave argmax (tie -> smaller index, matching argmax-first)
#pragma unroll
        for (int off = 16; off > 0; off >>= 1) {
            float od = __shfl_xor(best, off, 32);
            int   oi = __shfl_xor(bestn, off, 32);
            if (od > best || (od == best && oi < bestn)) { best = od; bestn = oi; }
        }
        if (lane == 0) { red_d[wid] = best; red_i[wid] = bestn; }
        __syncthreads();
        if (wid == 0) {
            best = red_d[lane]; bestn = red_i[lane];
#pragma unroll
            for (int off = 16; off > 0; off >>= 1) {
                float od = __shfl_xor(best, off, 32);
                int   oi = __shfl_xor(bestn, off, 32);
                if (od > best || (od == best && oi < bestn)) { best = od; bestn = oi; }
            }
            if (lane == 0) {
                bwin = bestn;
                out_ind[(size_t)b * MPTS + i] = (float)bestn;
            }
        }
        __syncthreads();
        int win = bwin;
        if ((win & (FPS_T - 1)) == tid) {              // owner thread publishes coords
            int r = win >> 10;
#pragma unroll
            for (int rr = 0; rr < FPS_R; ++rr)
                if (rr == r) {
                    bx = px[rr]; by = py[rr]; bz = pz[rr];
                    ws_sxyz[((size_t)b * 3 + 0) * MPTS + i] = px[rr];
                    ws_sxyz[((size_t)b * 3 + 1) * MPTS + i] = py[rr];
                    ws_sxyz[((size_t)b * 3 + 2) * MPTS + i] = pz[rr];
                }
        }
        __syncthreads();
    }
}

// ---------------------------------------------------------------------------
// Kernel B: kNN via V_WMMA_F32_16X16X4_F32. One wave owns a 16-sample tile and
// sweeps all 16384 points, 64 columns (4 WMMAs) per trip. D tiles bounce
// through LDS once to transpose ownership so each lane owns one output row for
// the register-resident sorted top-16 (key = p2[n] - 2*dot).
// ---------------------------------------------------------------------------
__global__ __launch_bounds__(32) void knn_kernel(const float* __restrict__ xyz,
                                                 const float* __restrict__ ws_sxyz,
                                                 const float* __restrict__ ws_p2,
                                                 int*   __restrict__ ws_nind,     // [B][M][K]
                                                 float* __restrict__ out_nind_f,  // [B][M][K]
                                                 float* __restrict__ out_sxyz) {  // [B][3][M]
    const int b = blockIdx.y;
    const int tile = blockIdx.x;                       // 0..127
    const int l = threadIdx.x;                         // 0..31
    const int mloc = l & 15;
    const bool hi = l >= 16;
    const int m = tile * 16 + mloc;

    const float* xb = xyz + (size_t)b * 3 * NPTS;
    const float* sx = ws_sxyz + (size_t)b * 3 * MPTS;
    const float* p2 = ws_p2 + (size_t)b * NPTS;
    const float4* p2v = (const float4*)p2;             // 16B-aligned (ws base)

    // A matrix (16x4 f32): lanes 0-15 hold K=0/1, lanes 16-31 hold K=2/3(=pad).
    // Branchless: unconditional loads, then mask — no exec save/restore.
    const int dimlo = hi ? 2 * NPTS : 0;               // row feeding VGPR0
    v2f a;
    a.x = sx[(hi ? 2 * MPTS : 0) + m];
    {
        float ay = sx[MPTS + m];
        a.y = hi ? 0.0f : ay;
    }

    float kd[KNN]; int kn[KNN];
#pragma unroll
    for (int j = 0; j < KNN; ++j) { kd[j] = BIGF; kn[j] = 0; }

    __shared__ float tileD[4][16][17];
    const int myrow = mloc;
    const int colbase = hi ? 8 : 0;
    const int rowoff = hi ? 8 : 0;

    for (int base = 0; base < NPTS; base += 64) {
        // ---- 4 back-to-back WMMAs (64 columns) ----
#pragma unroll
        for (int u = 0; u < 4; ++u) {
            int n = base + u * 16 + mloc;
            v2f bb;
            bb.x = xb[dimlo + n];                      // unconditional
            float yv = xb[NPTS + n];                   // unconditional
            bb.y = hi ? 0.0f : yv;                     // cndmask, no branch

            v8f c = {};
            c = __builtin_amdgcn_wmma_f32_16x16x4_f32(
                    /*neg_a=*/false, a, /*neg_b=*/false, bb,
                    /*c_mod=*/(short)0, c, /*reuse_a=*/false, /*reuse_b=*/false);

            // D vgpr r: lanes<16 -> row r, lanes>=16 -> row r+8
#pragma unroll
            for (int r = 0; r < 8; ++r) tileD[u][r + rowoff][mloc] = c[r];
        }
        __builtin_prefetch(xb + base + 1024, 0, 0);    // next chunks -> caches
        __syncthreads();

        // ---- selection: lane owns (row = mloc, 8 columns) per sub-tile ----
#pragma unroll
        for (int u = 0; u < 4; ++u) {
            int nb = base + u * 16 + colbase;
            float4 pA = p2v[nb >> 2];                  // p2[nb .. nb+3]
            float4 pB = p2v[(nb >> 2) + 1];            // p2[nb+4 .. nb+7]
            float pv[8] = {pA.x, pA.y, pA.z, pA.w, pB.x, pB.y, pB.z, pB.w};
#pragma unroll
            for (int j = 0; j < 8; ++j) {
                float d = pv[j] - 2.0f * tileD[u][myrow][colbase + j];
                int nn = nb + j;
                if (d < kd[KNN - 1]) {                 // cheap reject
#pragma unroll
                    for (int t = KNN - 1; t >= 0; --t) {
                        float pd = (t > 0) ? kd[t - 1] : -BIGF;
                        int   pn = (t > 0) ? kn[t - 1] : 0;
                        bool keep  = (kd[t] <= d);
                        bool shift = (pd > d);
                        kd[t] = keep ? kd[t] : (shift ? pd : d);
                        kn[t] = keep ? kn[t] : (shift ? pn : nn);
                    }
                }
            }
        }
        __syncthreads();
    }

    // merge the two per-row half-lists (sorted, 16+16 -> 16 smallest)
    __shared__ float md[16][32];
    __shared__ int   mi_[16][32];
#pragma unroll
    for (int j = 0; j < KNN; ++j) {
        md[myrow][(hi ? 16 : 0) + j]  = kd[j];
        mi_[myrow][(hi ? 16 : 0) + j] = kn[j];
    }
    __syncthreads();

    if (!hi) {
        int row = mloc;
        int pa = 0, pb = 16;
        int nk[KNN];
#pragma unroll
        for (int t = 0; t < KNN; ++t) {
            float da = (pa < 16) ? md[row][pa] : BIGF;
            float db = (pb < 32) ? md[row][pb] : BIGF;
            if (da <= db) { nk[t] = mi_[row][pa]; ++pa; }
            else          { nk[t] = mi_[row][pb]; ++pb; }
        }
        float sx0 = 0.f, sy0 = 0.f, sz0 = 0.f;
#pragma unroll
        for (int t = 0; t < KNN; ++t) {
            int n = nk[t];
            size_t o = ((size_t)b * MPTS + m) * KNN + t;
            ws_nind[o] = n;
            out_nind_f[o] = (float)n;
            sx0 += xb[n];
            sy0 += xb[NPTS + n];
            sz0 += xb[2 * NPTS + n];
        }
        out_sxyz[((size_t)b * 3 + 0) * MPTS + m] = sx0 * (1.0f / KNN);
        out_sxyz[((size_t)b * 3 + 1) * MPTS + m] = sy0 * (1.0f / KNN);
        out_sxyz[((size_t)b * 3 + 2) * MPTS + m] = sz0 * (1.0f / KNN);
    }
}

// ---------------------------------------------------------------------------
// Kernel C: feature gather + max over K. Block = 256 threads = 4 samples x 64
// features; lane accesses strided across features, x rows stay L2-resident (32MB << 192MB).
// ---------------------------------------------------------------------------
__global__ __launch_bounds__(256) void gather_max_kernel(const float* __restrict__ x,
                                                         const int* __restrict__ ws_nind,
                                                         float* __restrict__ out_x) {
    const int b = blockIdx.y;
    const int mi = threadIdx.x >> 6;     // 0..3
    const int d  = threadIdx.x & 63;
    const int m  = blockIdx.x * 4 + mi;

    __shared__ int inds[4][KNN];
    if (d < KNN) inds[mi][d] = ws_nind[((size_t)b * MPTS + m) * KNN + d];
    __syncthreads();

    const float* row = x + ((size_t)(b * D_FEAT + d)) * NPTS;
    float v = -BIGF;
#pragma unroll
    for (int k = 0; k < KNN; ++k) v = fmaxf(v, row[inds[mi][k]]);
    out_x[((size_t)(b * D_FEAT + d)) * MPTS + m] = v;
}

// ---------------------------------------------------------------------------
// Launcher
// ---------------------------------------------------------------------------
extern "C" void kernel_launch(void* const* d_in, const int* in_sizes, int n_in,
                              void* d_out, int out_size, void* d_ws, size_t ws_size,
                              hipStream_t stream) {
    const float* x   = (const float*)d_in[0];   // [8,64,16384]
    const float* xyz = (const float*)d_in[1];   // [8,3,16384]

    // workspace: p2 [B*N] | sample_xyz [B*3*M] | neighbor_ind(int) [B*M*K]
    float* ws_p2   = (float*)d_ws;
    float* ws_sxyz = ws_p2 + (size_t)B_SZ * NPTS;
    int*   ws_nind = (int*)(ws_sxyz + (size_t)B_SZ * 3 * MPTS);

    // outputs flat in return order
    float* out_x    = (float*)d_out;                         // [8,64,2048]
    float* out_sxyz = out_x + (size_t)B_SZ * D_FEAT * MPTS;  // [8,3,2048]
    float* out_ind  = out_sxyz + (size_t)B_SZ * 3 * MPTS;    // [8,2048]
    float* out_nind = out_ind + (size_t)B_SZ * MPTS;         // [8,2048,16]

    p2_kernel<<<(B_SZ * NPTS + 255) / 256, 256, 0, stream>>>(xyz, ws_p2);
    fps_kernel<<<B_SZ, FPS_T, 0, stream>>>(xyz, out_ind, ws_sxyz);
    knn_kernel<<<dim3(MPTS / 16, B_SZ), 32, 0, stream>>>(xyz, ws_sxyz, ws_p2,
                                                         ws_nind, out_nind, out_sxyz);
    gather_max_kernel<<<dim3(MPTS / 4, B_SZ), 256, 0, stream>>>(x, ws_nind, out_x);
}